// MultiHeadActor_69466801045946
// MI455X (gfx1250) — compile-verified
//
#include <hip/hip_runtime.h>
#include <math.h>

typedef float v2f __attribute__((ext_vector_type(2)));
typedef float v4f __attribute__((ext_vector_type(4)));
typedef float v8f __attribute__((ext_vector_type(8)));

#define NHEADS 16
#define HDIM   256
#define OBSD   64
#define NACT   12
#define BT     64          // batch rows per block (4 row tiles)
#define ASTR   260         // activation LDS row stride (bank-staggered, 16B aligned)
#define OSTR   36          // out tile LDS row stride

// ---- meta layout in d_ws (ints) ----
// [0..15]  counts   [16..32] row offsets (17)
// [33..49] tile offsets (17)   [50..65] cursors   [96..] rows[B]
#define M_COUNT 0
#define M_OFF   16
#define M_TOFF  33
#define M_CUR   50
#define M_ROWS  96

__global__ void k_zero(int* meta) { meta[threadIdx.x] = 0; }

__global__ void k_count(const int* __restrict__ idx, int* __restrict__ meta, int B) {
  int b = blockIdx.x * blockDim.x + threadIdx.x;
  if (b < B) atomicAdd(&meta[M_COUNT + idx[b]], 1);
}

__global__ void k_prefix(int* meta) {
  if (threadIdx.x == 0) {
    int off = 0, toff = 0;
    meta[M_OFF] = 0;
    meta[M_TOFF] = 0;
    for (int k = 0; k < NHEADS; ++k) {
      int c = meta[M_COUNT + k];
      off  += c;
      toff += (c + BT - 1) / BT;
      meta[M_OFF + k + 1]  = off;
      meta[M_TOFF + k + 1] = toff;
      meta[M_CUR + k] = 0;
    }
  }
}

__global__ void k_scatter(const int* __restrict__ idx, int* __restrict__ meta, int B) {
  int b = blockIdx.x * blockDim.x + threadIdx.x;
  if (b < B) {
    int k = idx[b];
    int pos = atomicAdd(&meta[M_CUR + k], 1);
    meta[M_ROWS + meta[M_OFF + k] + pos] = b;
  }
}

#define WMMA_F32(a, b, c) __builtin_amdgcn_wmma_f32_16x16x4_f32( \
    false, (a), false, (b), (short)0, (c), false, false)

// 4x2 register-blocked 16x16 tiles: 64 rows (A from LDS) x 32 cols (B from a
// row-major global weight [Kdim x ldw]). A frags shared across the 2 col tiles,
// B frags shared across the 4 row tiles -> 0.5 global dword + 1 LDS dword per
// lane per wmma. Column clamped to maxcol so dead lanes never read OOB.
__device__ __forceinline__ void wmma_gemm_4x2(
    const float* __restrict__ Wg, int ldw, int maxcol,
    const float* ldsA, int Kdim, int n0, int lane,
    v8f (&acc)[4][2])
{
  const int n   = lane & 15;              // N col (B/C/D lane index) == A row M
  const int kup = (lane >> 4) << 1;       // upper half-wave handles K+2/K+3
  const int col0 = min(n0 + n, maxcol);
  const int col1 = min(n0 + 16 + n, maxcol);
  const float* bp0 = Wg + (size_t)kup * ldw + col0;
  const float* bp1 = Wg + (size_t)kup * ldw + col1;
  const float* ap  = ldsA + n * ASTR + kup;    // row tile r at +16*r*ASTR
#pragma unroll 2
  for (int k0 = 0; k0 < Kdim; k0 += 4) {
    v2f a[4];
#pragma unroll
    for (int r = 0; r < 4; ++r)
      a[r] = *(const v2f*)(ap + r * (16 * ASTR));
    v2f b0, b1;
    b0.x = bp0[0]; b0.y = bp0[ldw];
    b1.x = bp1[0]; b1.y = bp1[ldw];
#pragma unroll
    for (int r = 0; r < 4; ++r) {
      acc[r][0] = WMMA_F32(a[r], b0, acc[r][0]);
      acc[r][1] = WMMA_F32(a[r], b1, acc[r][1]);
    }
    ap  += 4;
    bp0 += (size_t)4 * ldw;
    bp1 += (size_t)4 * ldw;
  }
}

// Single 16x16 tile (layer 4).
__device__ __forceinline__ void wmma_gemm_1(
    const float* __restrict__ Wg, int ldw, int maxcol,
    const float* ldsA, int Kdim, int n0, int lane, v8f& acc)
{
  const int n   = lane & 15;
  const int kup = (lane >> 4) << 1;
  const int col = min(n0 + n, maxcol);
  const float* bp = Wg + (size_t)kup * ldw + col;
  const float* ap = ldsA + n * ASTR + kup;
#pragma unroll 8
  for (int k0 = 0; k0 < Kdim; k0 += 4) {
    v2f a = *(const v2f*)ap;
    v2f b;
    b.x = bp[0]; b.y = bp[ldw];
    acc = WMMA_F32(a, b, acc);
    ap += 4;
    bp += (size_t)4 * ldw;
  }
}

// C/D layout: VGPR i -> row i (lanes 0-15) / row i+8 (lanes 16-31), col = lane&15.
__device__ __forceinline__ void store_act(
    float* lds_out, int ostride, int n0, int lane, v8f acc,
    const float* __restrict__ bias, int maxcol, bool relu)
{
  const int n = n0 + (lane & 15);
  const float bv = bias[min(n, maxcol)];
  const int rbase = (lane >> 4) << 3;
#pragma unroll
  for (int i = 0; i < 8; ++i) {
    float v = acc[i] + bv;
    if (relu) v = fmaxf(v, 0.0f);
    lds_out[(rbase + i) * ostride + n] = v;
  }
}

// One fused layer: dst = act(src @ Wg + bias), 64 rows, 32-col slab per wave.
__device__ __forceinline__ void layer_4x2(
    const float* __restrict__ Wg, const float* __restrict__ bias,
    const float* src, float* dst, int Kdim, int n0, int lane, bool relu)
{
  v8f acc[4][2] = {};
  wmma_gemm_4x2(Wg, HDIM, HDIM - 1, src, Kdim, n0, lane, acc);
#pragma unroll
  for (int r = 0; r < 4; ++r) {
    store_act(dst + r * (16 * ASTR), ASTR, n0,      lane, acc[r][0], bias, HDIM - 1, relu);
    store_act(dst + r * (16 * ASTR), ASTR, n0 + 16, lane, acc[r][1], bias, HDIM - 1, relu);
  }
}

__global__ __launch_bounds__(256) void k_main(
    const float* __restrict__ obs, const float* __restrict__ noise,
    const float* __restrict__ W1, const float* __restrict__ b1,
    const float* __restrict__ W2, const float* __restrict__ b2,
    const float* __restrict__ HW1, const float* __restrict__ Hb1,
    const float* __restrict__ HW2, const float* __restrict__ Hb2,
    const int* __restrict__ meta, float* __restrict__ out, int B)
{
  extern __shared__ float smem[];
  float* bufA = smem;                         // [BT][ASTR]
  float* bufB = smem + BT * ASTR;             // [BT][ASTR]
  float* outS = smem + 2 * BT * ASTR;         // [BT][OSTR]
  int*   ridS = (int*)(smem + 2 * BT * ASTR + BT * OSTR);  // [BT]

  const int t = blockIdx.x;
  const int totalTiles = meta[M_TOFF + NHEADS];
  if (t >= totalTiles) return;

  // map tile -> (head k, local tile)
  int k = 0;
  while (k < NHEADS - 1 && t >= meta[M_TOFF + k + 1]) ++k;
  const int lt       = t - meta[M_TOFF + k];
  const int binStart = meta[M_OFF + k];
  const int binEnd   = meta[M_OFF + k + 1];
  const int rstart   = binStart + lt * BT;
  const int nrows    = min(BT, binEnd - rstart);

  const int tid  = threadIdx.x;
  const int lane = tid & 31;
  const int wave = tid >> 5;

  if (tid < BT) ridS[tid] = meta[M_ROWS + rstart + min(tid, nrows - 1)];
  __syncthreads();

  { // gather 64 obs rows (same head) into bufA: 4 float4 per thread
    const int i = tid >> 2;                 // row 0..63
    const int j = (tid & 3) * 16;           // float col base (4 float4s)
    const float* src = obs + (size_t)ridS[i] * OBSD + j;
    float* dst = bufA + i * ASTR + j;
#pragma unroll
    for (int q = 0; q < 4; ++q)
      *(v4f*)(dst + 4 * q) = *(const v4f*)(src + 4 * q);
  }
  __syncthreads();

  const int n0 = wave * 32;   // each wave owns a 32-wide slab of H

  // layer 1: h1 = relu(obs @ W1 + b1)                 bufA -> bufB
  layer_4x2(W1, b1, bufA, bufB, OBSD, n0, lane, true);
  __syncthreads();

  // layer 2: h2 = relu(h1 @ W2 + b2)                  bufB -> bufA
  layer_4x2(W2, b2, bufB, bufA, HDIM, n0, lane, true);
  __syncthreads();

  // layer 3, selected head: y = relu(h2 @ HW1[k] + Hb1[k])   bufA -> bufB
  layer_4x2(HW1 + (size_t)k * HDIM * HDIM, Hb1 + (size_t)k * HDIM,
            bufA, bufB, HDIM, n0, lane, true);
  __syncthreads();

  // layer 4: out = y @ HW2[k] + Hb2[k]  ([64x256] x [256x24]); 8 tiles, all waves
  {
    const float* HW2p = HW2 + (size_t)k * HDIM * 2 * NACT;
    const float* Hb2p = Hb2 + (size_t)k * 2 * NACT;
    const int rt = wave >> 1;             // row tile 0..3
    const int ct = wave & 1;              // col tile 0/1
    v8f acc = {};
    wmma_gemm_1(HW2p, 2 * NACT, 2 * NACT - 1, bufB + rt * (16 * ASTR),
                HDIM, ct * 16, lane, acc);
    store_act(outS + rt * (16 * OSTR), OSTR, ct * 16, lane, acc,
              Hb2p, 2 * NACT - 1, false);
  }
  __syncthreads();

  // epilogue: (act-mean)/std == noise, so logp = -0.5*noise^2 - log_std - c
  if (tid < nrows) {
    const int r = ridS[tid];
    const float HALF_LOG_2PI = 0.9189385332046727f;
    float slogp = 0.0f;
#pragma unroll
    for (int a = 0; a < NACT; ++a) {
      float mean    = outS[tid * OSTR + a];
      float ls      = outS[tid * OSTR + NACT + a];
      float log_std = -10.0f + 6.0f * (tanhf(ls) + 1.0f);
      float sd      = expf(log_std);
      float nz      = noise[(size_t)r * NACT + a];
      float act     = mean + sd * nz;
      float sq      = tanhf(act);
      slogp += (-0.5f * nz * nz - log_std - HALF_LOG_2PI)
             - logf(1.0f - sq * sq + 1e-6f);
      out[(size_t)r * NACT + a] = sq;
    }
    out[(size_t)B * NACT + r] = slogp;
  }
}

extern "C" void kernel_launch(void* const* d_in, const int* in_sizes, int n_in,
                              void* d_out, int out_size, void* d_ws, size_t ws_size,
                              hipStream_t stream) {
  const float* obs   = (const float*)d_in[0];
  const int*   idx   = (const int*)  d_in[1];
  const float* noise = (const float*)d_in[2];
  const float* W1    = (const float*)d_in[3];
  const float* b1    = (const float*)d_in[4];
  const float* W2    = (const float*)d_in[5];
  const float* b2    = (const float*)d_in[6];
  const float* HW1   = (const float*)d_in[7];
  const float* Hb1   = (const float*)d_in[8];
  const float* HW2   = (const float*)d_in[9];
  const float* Hb2   = (const float*)d_in[10];
  const int B = in_sizes[1];                 // idx has one entry per batch row

  int* meta = (int*)d_ws;                    // meta + binned row ids live in ws

  k_zero   <<<1, 96, 0, stream>>>(meta);
  k_count  <<<(B + 255) / 256, 256, 0, stream>>>(idx, meta, B);
  k_prefix <<<1, 32, 0, stream>>>(meta);
  k_scatter<<<(B + 255) / 256, 256, 0, stream>>>(idx, meta, B);

  const int maxTiles = (B + BT - 1) / BT + NHEADS;  // exact count read on device
  const size_t smemBytes = (size_t)(2 * BT * ASTR + BT * OSTR) * sizeof(float)
                         + BT * sizeof(int);        // ~139 KB (< 320 KB/WGP, 2 blk/WGP)
  k_main<<<maxTiles, 256, smemBytes, stream>>>(obs, noise, W1, b1, W2, b2,
                                               HW1, Hb1, HW2, Hb2,
                                               meta, (float*)d_out, B);
}